// EdgeGATConv_63685775065179
// MI455X (gfx1250) — compile-verified
//
#include <hip/hip_runtime.h>

#define NNODES 50000
#define NEDGES 800000
#define IN_DIM 128
#define EF_DIM 64
#define NH 4
#define NF 16
#define HF 64   // NH*NF

#define EDGE_TILES_PER_BLOCK 10
#define EDGE_BLOCKS (NEDGES / 16 / EDGE_TILES_PER_BLOCK)   // 5000

typedef __attribute__((ext_vector_type(2))) float v2f;
typedef __attribute__((ext_vector_type(8))) float v8f;

// sign-aware float atomic max via int/uint atomics (works for mixed signs)
__device__ __forceinline__ void atomicMaxF(float* addr, float val) {
    if (val >= 0.0f) {
        atomicMax((int*)addr, __float_as_int(val));
    } else {
        atomicMin((unsigned int*)addr, __float_as_uint(val));
    }
}

__global__ void init_kernel(float* __restrict__ emax, float* __restrict__ denom, int n) {
    int i = blockIdx.x * blockDim.x + threadIdx.x;
    if (i < n) { emax[i] = -3.0e38f; denom[i] = 0.0f; }
}

// ------------------------------------------------------------------
// Node GEMM: feat[N,128] x {W_fc, W_res}[128,64] with fp32 WMMA.
// Block = 128 threads (4 waves). Block handles 16 rows; wave w = head w
// = columns 16w..16w+15. Fuses el/er head reductions and out = res + bias.
// ------------------------------------------------------------------
__global__ __launch_bounds__(128) void node_gemm_kernel(
    const float* __restrict__ feat, const float* __restrict__ W_fc,
    const float* __restrict__ W_res, const float* __restrict__ attn_l,
    const float* __restrict__ attn_r, const float* __restrict__ bias,
    float* __restrict__ feat_src, float* __restrict__ el, float* __restrict__ er,
    float* __restrict__ out) {
    __shared__ float As[16][IN_DIM + 4];   // +4 pad: rows land on distinct banks
    const int tid  = threadIdx.x;
    const int row0 = blockIdx.x * 16;

    for (int i = tid; i < 16 * (IN_DIM / 4); i += 128) {
        int r = i / (IN_DIM / 4);
        int c = (i % (IN_DIM / 4)) * 4;
        *(float4*)&As[r][c] = *(const float4*)(feat + (size_t)(row0 + r) * IN_DIM + c);
    }
    __syncthreads();

    const int wave  = tid >> 5;        // head / 16-col tile
    const int lane  = tid & 31;
    const int l15   = lane & 15;
    const int khalf = lane >> 4;
    const int col   = (wave << 4) + l15;   // 0..63

    v8f c_fc  = {};
    v8f c_res = {};
    for (int k0 = 0; k0 < IN_DIM; k0 += 4) {
        const int k = k0 + khalf * 2;
        v2f a;  a.x  = As[l15][k];            a.y  = As[l15][k + 1];
        v2f b0; b0.x = W_fc[k * HF + col];    b0.y = W_fc[(k + 1) * HF + col];
        v2f b1; b1.x = W_res[k * HF + col];   b1.y = W_res[(k + 1) * HF + col];
        c_fc  = __builtin_amdgcn_wmma_f32_16x16x4_f32(false, a, false, b0, (short)0, c_fc,  false, false);
        c_res = __builtin_amdgcn_wmma_f32_16x16x4_f32(false, a, false, b1, (short)0, c_res, false, false);
    }

    const float al = attn_l[col];   // attn_l flat [H*F], col == h*16+f
    const float ar = attn_r[col];
    const float bs = bias[col];
#pragma unroll
    for (int i = 0; i < 8; ++i) {
        const int r = row0 + i + khalf * 8;      // C-tile row for this lane half
        feat_src[(size_t)r * HF + col] = c_fc[i];
        out[(size_t)r * HF + col]      = c_res[i] + bs;
        float pl = c_fc[i] * al;
        float pr = c_fc[i] * ar;
#pragma unroll
        for (int m = 8; m >= 1; m >>= 1) {       // reduce over f (16 lanes/half)
            pl += __shfl_xor(pl, m, 32);
            pr += __shfl_xor(pr, m, 32);
        }
        if (l15 == 0) {
            el[r * NH + wave] = pl;
            er[r * NH + wave] = pr;
        }
    }
}

// ------------------------------------------------------------------
// Edge GEMM: edge_feat[E,64] x W_edge[64,64] with fp32 WMMA.
// Blocked: each block owns EDGE_TILES_PER_BLOCK consecutive tiles of 16 edges.
// Each wave preloads its full B slice (16 v2f regs) ONCE, so the inner
// loop is pure LDS-read + v_wmma. Prefetches the next A tile.
// Fuses ee reduction, e = leakyrelu(el[src]+er[dst]+ee), segment-max.
// ------------------------------------------------------------------
__global__ __launch_bounds__(128) void edge_gemm_kernel(
    const float* __restrict__ edge_feat, const float* __restrict__ W_edge,
    const float* __restrict__ attn_e, const float* __restrict__ el,
    const float* __restrict__ er, const int* __restrict__ src,
    const int* __restrict__ dst, float* __restrict__ feat_e,
    float* __restrict__ eLogit, float* __restrict__ emax) {
    __shared__ float As[16][EF_DIM + 4];
    const int tid   = threadIdx.x;
    const int wave  = tid >> 5;
    const int lane  = tid & 31;
    const int l15   = lane & 15;
    const int khalf = lane >> 4;
    const int col   = (wave << 4) + l15;

    // Preload the wave's B slice of W_edge: k pairs for all 16 k-steps.
    v2f bReg[16];
#pragma unroll
    for (int i = 0; i < 16; ++i) {
        const int k = i * 4 + khalf * 2;
        bReg[i].x = W_edge[k * HF + col];
        bReg[i].y = W_edge[(k + 1) * HF + col];
    }
    const float ae = attn_e[col];

    for (int t = 0; t < EDGE_TILES_PER_BLOCK; ++t) {
        const int row0 = (blockIdx.x * EDGE_TILES_PER_BLOCK + t) * 16;

        // Prefetch next tile's A while we work on this one (global_prefetch_b8).
        if (t + 1 < EDGE_TILES_PER_BLOCK) {
            __builtin_prefetch(edge_feat + (size_t)(row0 + 16) * EF_DIM + tid * 8, 0, 3);
        }

        // Stage 16x64 A tile in LDS: 1024 floats = 128 threads x 2 float4.
        for (int i = tid; i < 16 * (EF_DIM / 4); i += 128) {
            int r = i / (EF_DIM / 4);
            int c = (i % (EF_DIM / 4)) * 4;
            *(float4*)&As[r][c] = *(const float4*)(edge_feat + (size_t)(row0 + r) * EF_DIM + c);
        }
        __syncthreads();

        v8f acc = {};
#pragma unroll
        for (int i = 0; i < 16; ++i) {
            const int k = i * 4 + khalf * 2;
            v2f a; a.x = As[l15][k]; a.y = As[l15][k + 1];
            acc = __builtin_amdgcn_wmma_f32_16x16x4_f32(false, a, false, bReg[i], (short)0, acc, false, false);
        }

#pragma unroll
        for (int i = 0; i < 8; ++i) {
            const int eidx = row0 + i + khalf * 8;
            feat_e[(size_t)eidx * HF + col] = acc[i];
            float pe = acc[i] * ae;
#pragma unroll
            for (int m = 8; m >= 1; m >>= 1) pe += __shfl_xor(pe, m, 32);
            if (l15 == 0) {
                const int s = src[eidx];
                const int d = dst[eidx];
                float ev = pe + el[s * NH + wave] + er[d * NH + wave];
                ev = ev > 0.0f ? ev : 0.2f * ev;            // leaky relu, slope 0.2
                eLogit[(size_t)eidx * NH + wave] = ev;
                atomicMaxF(&emax[d * NH + wave], ev);
            }
        }
        __syncthreads();   // all waves done reading As before next tile load
    }
}

// ex = exp(e - emax[dst]); denom[dst] += ex. One thread per (edge, head).
__global__ void softmax_denom_kernel(const int* __restrict__ dst,
                                     const float* __restrict__ emax,
                                     float* __restrict__ eLogit,
                                     float* __restrict__ denom) {
    const int idx = blockIdx.x * 256 + threadIdx.x;   // grid sized exactly E*H
    const int h = idx & (NH - 1);
    const int e = idx >> 2;
    const int d = dst[e];
    const float ex = __expf(eLogit[idx] - emax[d * NH + h]);
    eLogit[idx] = ex;                                  // overwrite logit with exp
    atomicAdd(&denom[d * NH + h], ex);
}

// msg = (feat_src[src] + feat_e) * a, scatter-add into out. Thread per (edge, c).
__global__ void scatter_kernel(const int* __restrict__ src, const int* __restrict__ dst,
                               const float* __restrict__ eLogit, const float* __restrict__ denom,
                               const float* __restrict__ feat_src, const float* __restrict__ feat_e,
                               float* __restrict__ out) {
    const size_t idx = (size_t)blockIdx.x * 256 + threadIdx.x;  // grid sized exactly E*64
    const int    c   = (int)(idx & 63);
    const size_t e   = idx >> 6;
    const int    h   = c >> 4;
    const int    s   = src[e];
    const int    d   = dst[e];
    const float  a   = eLogit[e * NH + h] / denom[d * NH + h];
    const float  m   = (feat_src[(size_t)s * HF + c] + feat_e[e * HF + c]) * a;
    atomicAdd(&out[(size_t)d * HF + c], m);
}

extern "C" void kernel_launch(void* const* d_in, const int* in_sizes, int n_in,
                              void* d_out, int out_size, void* d_ws, size_t ws_size,
                              hipStream_t stream) {
    const float* feat      = (const float*)d_in[0];
    const float* edge_feat = (const float*)d_in[1];
    const float* W_fc      = (const float*)d_in[2];
    const float* W_edge    = (const float*)d_in[3];
    const float* attn_l    = (const float*)d_in[4];
    const float* attn_r    = (const float*)d_in[5];
    const float* attn_e    = (const float*)d_in[6];
    const float* bias      = (const float*)d_in[7];
    const float* W_res     = (const float*)d_in[8];
    const int*   src       = (const int*)d_in[9];
    const int*   dst       = (const int*)d_in[10];
    float*       out       = (float*)d_out;

    float* ws = (float*)d_ws;
    float* feat_src = ws;  ws += (size_t)NNODES * HF;   // 3.2M f
    float* el       = ws;  ws += (size_t)NNODES * NH;   // 0.2M f
    float* er       = ws;  ws += (size_t)NNODES * NH;
    float* emax     = ws;  ws += (size_t)NNODES * NH;
    float* denom    = ws;  ws += (size_t)NNODES * NH;
    float* eLogit   = ws;  ws += (size_t)NEDGES * NH;   // 3.2M f
    float* feat_e   = ws;                               // 51.2M f

    init_kernel<<<(NNODES * NH + 255) / 256, 256, 0, stream>>>(emax, denom, NNODES * NH);
    node_gemm_kernel<<<NNODES / 16, 128, 0, stream>>>(feat, W_fc, W_res, attn_l, attn_r,
                                                      bias, feat_src, el, er, out);
    edge_gemm_kernel<<<EDGE_BLOCKS, 128, 0, stream>>>(edge_feat, W_edge, attn_e, el, er,
                                                      src, dst, feat_e, eLogit, emax);
    softmax_denom_kernel<<<(NEDGES * NH) / 256, 256, 0, stream>>>(dst, emax, eLogit, denom);
    scatter_kernel<<<(NEDGES * 64) / 256, 256, 0, stream>>>(src, dst, eLogit, denom,
                                                            feat_src, feat_e, out);
}